// BuiltParametrizedOperator_31662498906404
// MI455X (gfx1250) — compile-verified
//
#include <hip/hip_runtime.h>

// RX(theta) on one qubit of a [B, 2^NQ] complex state (real/imag split f32).
// Memory-bound streaming op: 536 MB read + 536 MB written exactly once
// -> ~46 us floor at 23.3 TB/s HBM. No matrix contraction exists, so WMMA /
// TDM add nothing; the MI455X-optimal path is 128-bit non-temporal global
// loads/stores under wave32 with ~2^23 threads in flight.
//
// Transcendentals (accurate libm cosf/sinf on theta/2) are hoisted into a
// 1-thread prologue kernel writing {c, s} to d_ws, keeping the streaming
// kernel's instruction footprint minimal.

typedef float v4f __attribute__((ext_vector_type(4)));

__global__ void rx_coeff_kernel(const float* __restrict__ theta,
                                float* __restrict__ coeff) {
    if (threadIdx.x == 0) {
        const float half = 0.5f * theta[0];
        coeff[0] = cosf(half);   // accurate libm path, runs once total
        coeff[1] = sinf(half);
    }
}

__global__ __launch_bounds__(256) void rx_gate_kernel(
    const float* __restrict__ sr,      // state_real, N elements
    const float* __restrict__ si,      // state_imag, N elements
    const float* __restrict__ coeff,   // {c, s} from prologue
    const int*   __restrict__ qubit,   // 1 element
    const int*   __restrict__ nqubits, // 1 element
    float*       __restrict__ out,     // [out_r (N) | out_i (N)]
    long long N)                       // B * 2^NQ
{
    const long long t = (long long)blockIdx.x * blockDim.x + threadIdx.x;

    // Uniform scalars -> s_load / SALU.
    const int q     = qubit[0];
    const int NQ    = nqubits[0];
    const int shift = NQ - q - 1;            // log2(right)
    const long long right = 1LL << shift;

    const float c = coeff[0];
    const float s = coeff[1];

    float* __restrict__ out_r = out;
    float* __restrict__ out_i = out + N;

    const long long npairs = N >> 1;

    if (shift >= 2) {
        // ---- vec4 path: each thread handles 4 consecutive r-positions ----
        const long long nvec = npairs >> 2;          // N/8 threads do work
        if (t >= nvec) return;

        const int        vshift = shift - 2;
        const long long  vmask  = (1LL << vshift) - 1;
        const long long  row    = t >> vshift;       // combined (b, left) row
        const long long  rchunk = t & vmask;
        const long long  base0  = (row << (shift + 1)) + (rchunk << 2);
        const long long  base1  = base0 + right;

        const v4f r0 = __builtin_nontemporal_load((const v4f*)(sr + base0));
        const v4f r1 = __builtin_nontemporal_load((const v4f*)(sr + base1));
        const v4f i0 = __builtin_nontemporal_load((const v4f*)(si + base0));
        const v4f i1 = __builtin_nontemporal_load((const v4f*)(si + base1));

        const v4f nr0 = c * r0 + s * i1;
        const v4f ni0 = c * i0 - s * r1;
        const v4f nr1 = c * r1 + s * i0;
        const v4f ni1 = c * i1 - s * r0;

        __builtin_nontemporal_store(nr0, (v4f*)(out_r + base0));
        __builtin_nontemporal_store(nr1, (v4f*)(out_r + base1));
        __builtin_nontemporal_store(ni0, (v4f*)(out_i + base0));
        __builtin_nontemporal_store(ni1, (v4f*)(out_i + base1));
    } else {
        // ---- uniform scalar fallback (right < 4); 4 pairs per thread ----
        const long long rmask = right - 1;
        #pragma unroll
        for (int k = 0; k < 4; ++k) {
            const long long p = (t << 2) + k;
            if (p >= npairs) return;
            const long long row   = p >> shift;
            const long long rr    = p & rmask;
            const long long base0 = (row << (shift + 1)) + rr;
            const long long base1 = base0 + right;

            const float r0 = __builtin_nontemporal_load(sr + base0);
            const float r1 = __builtin_nontemporal_load(sr + base1);
            const float i0 = __builtin_nontemporal_load(si + base0);
            const float i1 = __builtin_nontemporal_load(si + base1);

            __builtin_nontemporal_store(fmaf(c, r0,  s * i1), out_r + base0);
            __builtin_nontemporal_store(fmaf(c, r1,  s * i0), out_r + base1);
            __builtin_nontemporal_store(fmaf(c, i0, -s * r1), out_i + base0);
            __builtin_nontemporal_store(fmaf(c, i1, -s * r0), out_i + base1);
        }
    }
}

extern "C" void kernel_launch(void* const* d_in, const int* in_sizes, int n_in,
                              void* d_out, int out_size, void* d_ws, size_t ws_size,
                              hipStream_t stream) {
    (void)n_in; (void)ws_size; (void)out_size;

    const float* sr    = (const float*)d_in[0];
    const float* si    = (const float*)d_in[1];
    const float* theta = (const float*)d_in[2];
    const int*   qb    = (const int*)d_in[3];
    const int*   nq    = (const int*)d_in[4];
    float*       out   = (float*)d_out;
    float*       coeff = (float*)d_ws;              // 2 floats of scratch

    const long long N = (long long)in_sizes[0];     // B * 2^NQ total amplitudes

    rx_coeff_kernel<<<1, 32, 0, stream>>>(theta, coeff);

    // One thread per 4 amplitude pairs (vec4 path) -> N/8 threads.
    const long long nthreads = (N + 7) >> 3;
    const int block = 256;
    const long long grid = (nthreads + block - 1) / block;

    rx_gate_kernel<<<dim3((unsigned)grid), dim3(block), 0, stream>>>(
        sr, si, coeff, qb, nq, out, N);
}